// CenterNetMulBranch_54288386622033
// MI455X (gfx1250) — compile-verified
//
#include <hip/hip_runtime.h>
#include <hip/hip_bf16.h>
#include <stdint.h>

#define N_  32
#define C_  80
#define H_  128
#define W_  128
#define HW_ (H_ * W_)
#define K_  100
#define BATCH_CAP 15360          // per-batch candidate capacity (80ch * 192)
#define S1_BINS 2048
#define S2_BINS 4096
#define NEG_INF (-3.0e38f)

#if defined(__HIP_DEVICE_COMPILE__) && __has_builtin(__builtin_amdgcn_global_load_async_to_lds_b128)
#define USE_ASYNC_LDS 1
#else
#define USE_ASYNC_LDS 0
#endif

#if USE_ASYNC_LDS
typedef int v4i_ __attribute__((ext_vector_type(4)));
typedef __attribute__((address_space(1))) v4i_ gv4i_;
typedef __attribute__((address_space(3))) v4i_ lv4i_;
#endif

__device__ __forceinline__ float sigmoidf_(float x) {
    return 1.0f / (1.0f + __expf(-x));
}

__device__ __forceinline__ void wait_async_lds() {
#if USE_ASYNC_LDS
#if __has_builtin(__builtin_amdgcn_s_wait_asynccnt)
    __builtin_amdgcn_s_wait_asynccnt(0);
#else
    asm volatile("s_wait_asynccnt 0" ::: "memory");
#endif
#endif
}

// ---------------------------------------------------------------------------
// Stage 1: one block per (n, c). Fused sigmoid + 3x3 NMS + per-channel
// histogram-thresholded top-~100 candidate compaction into per-batch buffer.
// ---------------------------------------------------------------------------
__global__ __launch_bounds__(256)
void s1_nms_select(const float* __restrict__ hm,
                   float* __restrict__ cscore,
                   int*   __restrict__ cidx,
                   int*   __restrict__ ccnt)
{
    __shared__ float    tile[HW_];        // 64 KB channel tile
    __shared__ unsigned hist[S1_BINS];    // 8 KB
    __shared__ unsigned seg[256];
    __shared__ int      scan[256];
    __shared__ int      s_thresh;
    __shared__ int      s_base;

    const int t = threadIdx.x;
    const int c = blockIdx.x;
    const int n = blockIdx.y;
    const float* src = hm + ((size_t)(n * C_ + c) << 14);

    // ---- stage the 128x128 tile into LDS (CDNA5 async path) ----
#if USE_ASYNC_LDS
    for (int i = 0; i < 16; ++i) {
        int j = t + (i << 8);   // float4 index, 4096 total
        __builtin_amdgcn_global_load_async_to_lds_b128(
            (gv4i_*)(src + (j << 2)),
            (lv4i_*)(&tile[j << 2]), 0, 0);
    }
#else
    {
        const float4* s4 = (const float4*)src;
        float4*       d4 = (float4*)tile;
        for (int i = 0; i < 16; ++i) { int j = t + (i << 8); d4[j] = s4[j]; }
    }
#endif
    for (int i = t; i < S1_BINS; i += 256) hist[i] = 0u;
    wait_async_lds();
    __syncthreads();

    // ---- 3x3 NMS with sliding column-max: thread owns 64 contiguous cols ----
    const int r  = t >> 1;
    const int cb = (t & 1) << 6;
    const float* rowp = &tile[r << 7];
    unsigned long long mask = 0ull;

    auto colmax = [&](int cc) -> float {
        if ((unsigned)cc > 127u) return NEG_INF;
        float m = rowp[cc];
        if (r > 0)       m = fmaxf(m, rowp[cc - 128]);
        if (r < H_ - 1)  m = fmaxf(m, rowp[cc + 128]);
        return m;
    };

    float cl = colmax(cb - 1);
    float cm = colmax(cb);
    for (int j = 0; j < 64; ++j) {
        float cr = colmax(cb + j + 1);
        float v  = rowp[cb + j];
        float m9 = fmaxf(cl, fmaxf(cm, cr));   // includes center
        if (v >= m9) {                          // pooled == hm  =>  keep
            mask |= (1ull << j);
            float s = sigmoidf_(v);
            int b = (int)(s * (float)S1_BINS);
            b = b < 0 ? 0 : (b > S1_BINS - 1 ? S1_BINS - 1 : b);
            atomicAdd(&hist[b], 1u);
        }
        cl = cm; cm = cr;
    }
    __syncthreads();

    // ---- find threshold bin: highest B with count(bin >= B) >= K ----
    unsigned segsum = 0;
    for (int b = 0; b < 8; ++b) segsum += hist[t * 8 + b];
    seg[t] = segsum;
    __syncthreads();
    if (t == 0) {
        int acc = 0, g = 255;
        for (; g >= 0; --g) { if (acc + (int)seg[g] >= K_) break; acc += (int)seg[g]; }
        int b = 0;
        if (g >= 0) {
            int acc2 = acc;
            for (b = g * 8 + 7; b >= g * 8; --b) { acc2 += (int)hist[b]; if (acc2 >= K_) break; }
            if (b < g * 8) b = g * 8;
        }
        s_thresh = b;
    }
    __syncthreads();
    const int B = s_thresh;

    // ---- count survivors above threshold, prune mask ----
    int cnt = 0;
    for (int j = 0; j < 64; ++j) {
        if ((mask >> j) & 1ull) {
            float s = sigmoidf_(rowp[cb + j]);
            int b = (int)(s * (float)S1_BINS);
            b = b < 0 ? 0 : (b > S1_BINS - 1 ? S1_BINS - 1 : b);
            if (b >= B) ++cnt; else mask &= ~(1ull << j);
        }
    }
    scan[t] = cnt;
    __syncthreads();
    for (int off = 1; off < 256; off <<= 1) {   // inclusive Hillis-Steele scan
        int v2 = (t >= off) ? scan[t - off] : 0;
        __syncthreads();
        scan[t] += v2;
        __syncthreads();
    }
    if (t == 0) s_base = atomicAdd(&ccnt[n], scan[255]);   // 1 atomic per block
    __syncthreads();
    int pos = s_base + scan[t] - cnt;

    // ---- compact (score, flat index) into per-batch candidate buffer ----
    float* csb = cscore + (size_t)n * BATCH_CAP;
    int*   cib = cidx   + (size_t)n * BATCH_CAP;
    for (int j = 0; j < 64; ++j) {
        if ((mask >> j) & 1ull) {
            if (pos < BATCH_CAP) {
                csb[pos] = sigmoidf_(rowp[cb + j]);
                cib[pos] = c * HW_ + (r << 7) + cb + j;
            }
            ++pos;
        }
    }
}

// ---------------------------------------------------------------------------
// Stage 2: one block per batch. Histogram-narrow to <=1024 candidates,
// bitonic sort (score desc, index asc), emit dets + categories.
// ---------------------------------------------------------------------------
__global__ __launch_bounds__(256)
void s2_topk_emit(const float* __restrict__ cscore,
                  const int*   __restrict__ cidx,
                  const int*   __restrict__ ccnt,
                  const float* __restrict__ offs,
                  const float* __restrict__ wh,
                  float* __restrict__ out)
{
    __shared__ unsigned           hist[S2_BINS];  // 16 KB
    __shared__ unsigned           seg[256];
    __shared__ unsigned long long keys[1024];     // 8 KB
    __shared__ int                s_thresh;
    __shared__ int                s_cnt;

    const int t = threadIdx.x;
    const int n = blockIdx.x;
    int M = ccnt[n]; if (M > BATCH_CAP) M = BATCH_CAP;
    const float* csb = cscore + (size_t)n * BATCH_CAP;
    const int*   cib = cidx   + (size_t)n * BATCH_CAP;

    for (int i = t; i < S2_BINS; i += 256) hist[i] = 0u;
    for (int q = t; q < 1024; q += 256) keys[q] = 0ull;
    if (t == 0) s_cnt = 0;
    __syncthreads();

    for (int i = t; i < M; i += 256) {
        int b = (int)(csb[i] * (float)S2_BINS);
        b = b < 0 ? 0 : (b > S2_BINS - 1 ? S2_BINS - 1 : b);
        atomicAdd(&hist[b], 1u);
    }
    __syncthreads();

    unsigned ss = 0;
    for (int b = 0; b < 16; ++b) ss += hist[t * 16 + b];
    seg[t] = ss;
    __syncthreads();
    if (t == 0) {
        int acc = 0, g = 255;
        for (; g >= 0; --g) { if (acc + (int)seg[g] >= K_) break; acc += (int)seg[g]; }
        int b = 0;
        if (g >= 0) {
            int acc2 = acc;
            for (b = g * 16 + 15; b >= g * 16; --b) { acc2 += (int)hist[b]; if (acc2 >= K_) break; }
            if (b < g * 16) b = g * 16;
        }
        s_thresh = b;
    }
    __syncthreads();
    const int B = s_thresh;

    // compact candidates >= threshold into monotonic sort keys
    for (int i = t; i < M; i += 256) {
        float s = csb[i];
        int b = (int)(s * (float)S2_BINS);
        b = b < 0 ? 0 : (b > S2_BINS - 1 ? S2_BINS - 1 : b);
        if (b >= B) {
            int q = atomicAdd(&s_cnt, 1);
            if (q < 1024) {
                unsigned u  = __float_as_uint(s);
                unsigned os = (u & 0x80000000u) ? ~u : (u | 0x80000000u);
                keys[q] = ((unsigned long long)os << 32)
                        | (unsigned long long)(0xFFFFFFFFu - (unsigned)cib[i]);
            }
        }
    }
    __syncthreads();

    // bitonic sort, descending (score desc; ties -> smaller flat index first)
    for (unsigned k = 2; k <= 1024; k <<= 1) {
        for (unsigned j = k >> 1; j > 0; j >>= 1) {
            for (unsigned q = t; q < 1024; q += 256) {
                unsigned ix = q ^ j;
                if (ix > q) {
                    unsigned long long a = keys[q], b2 = keys[ix];
                    bool desc = ((q & k) == 0);
                    if (desc ? (a < b2) : (a > b2)) { keys[q] = b2; keys[ix] = a; }
                }
            }
            __syncthreads();
        }
    }

    // emit dets (N,K,5) then categories (N,K) appended flat
    for (int rI = t; rI < K_; rI += 256) {
        unsigned long long key = keys[rI];
        float sc = 0.0f; int id = 0;
        if (key != 0ull) {
            unsigned os = (unsigned)(key >> 32);
            unsigned u  = (os & 0x80000000u) ? (os & 0x7FFFFFFFu) : ~os;
            sc = __uint_as_float(u);
            id = (int)(0xFFFFFFFFu - (unsigned)(key & 0xFFFFFFFFull));
        }
        int cat = id >> 14;            // id / (H*W)
        int sp  = id & (HW_ - 1);
        float ys = (float)(sp >> 7);
        float xs = (float)(sp & (W_ - 1));
        const float* op = offs + ((size_t)n * 2) * HW_;
        const float* wp = wh   + ((size_t)n * 2) * HW_;
        float o0 = op[sp], o1 = op[HW_ + sp];
        float bw = wp[sp], bh = wp[HW_ + sp];
        float cx = (xs + o0) * 4.0f;   // DOWN_STRIDE
        float cy = (ys + o1) * 4.0f;
        float* drow = out + (size_t)(n * K_ + rI) * 5;
        drow[0] = cx - bw * 0.5f;
        drow[1] = cy - bh * 0.5f;
        drow[2] = cx + bw * 0.5f;
        drow[3] = cy + bh * 0.5f;
        drow[4] = sc;
        out[(size_t)N_ * K_ * 5 + n * K_ + rI] = (float)cat;
    }
}

// ---------------------------------------------------------------------------
extern "C" void kernel_launch(void* const* d_in, const int* in_sizes, int n_in,
                              void* d_out, int out_size, void* d_ws, size_t ws_size,
                              hipStream_t stream)
{
    const float* hm   = (const float*)d_in[0];
    const float* offs = (const float*)d_in[1];
    const float* wh   = (const float*)d_in[2];
    float* out = (float*)d_out;

    int*   ccnt   = (int*)d_ws;
    float* cscore = (float*)((char*)d_ws + 256);
    int*   cidx   = (int*)((char*)d_ws + 256 + (size_t)N_ * BATCH_CAP * sizeof(float));

    (void)hipMemsetAsync(d_ws, 0, 256, stream);   // zero per-batch counters each call

    dim3 g1(C_, N_);
    s1_nms_select<<<g1, 256, 0, stream>>>(hm, cscore, cidx, ccnt);
    s2_topk_emit<<<N_, 256, 0, stream>>>(cscore, cidx, ccnt, offs, wh, out);
}